// GATLayer_15822659518635
// MI455X (gfx1250) — compile-verified
//
#include <hip/hip_runtime.h>
#include <hip/hip_bf16.h>
#include <math.h>

// ---------------- problem constants ----------------
#define N_NODES   50000
#define N_EDGES   800000
#define IN_F      128
#define HEADS     8
#define HEAD_DIM  16
#define OUT_F     (HEADS * HEAD_DIM)   // 128
#define ALPHA     0.2f
#define EPS_F     1e-6f

typedef __attribute__((ext_vector_type(2))) float v2f;
typedef __attribute__((ext_vector_type(8))) float v8f;

// ----- ordered-uint encoding for float atomicMax (handles negatives) -----
__device__ __forceinline__ unsigned enc_f32(float f) {
    unsigned u = __float_as_uint(f);
    return (u & 0x80000000u) ? ~u : (u | 0x80000000u);
}
__device__ __forceinline__ float dec_f32(unsigned u) {
    return (u & 0x80000000u) ? __uint_as_float(u ^ 0x80000000u)
                             : __uint_as_float(~u);
}
__device__ __forceinline__ float leaky(float s) {
    return s >= 0.0f ? s : ALPHA * s;
}

// ---------------------------------------------------------------
// 0) init: out = bias broadcast, denom = 0, gmax = enc(-inf)=0
// (must run every launch: atomics accumulate into these buffers)
// ---------------------------------------------------------------
__global__ void gat_init(float* __restrict__ out, const float* __restrict__ bias,
                         float* __restrict__ denom, unsigned* __restrict__ gmax) {
    const int i = blockIdx.x * blockDim.x + threadIdx.x;
    if (i < N_NODES * OUT_F) out[i] = bias[i & (OUT_F - 1)];
    if (i < N_NODES * HEADS) denom[i] = 0.0f;
    if (i < HEADS)           gmax[i] = 0u;   // 0 < enc(any finite float)
}

// ---------------------------------------------------------------
// 1) h = x @ Wcat  via V_WMMA_F32_16X16X4_F32
//    block: 256 thr = 8 waves; block tile = 16 rows x 128 cols;
//    wave w owns head w (cols w*16 .. w*16+15). K loop: 32 x K=4.
//    A-frag (ISA 16x4 f32 layout): lanes0-15 hold K=k,k+1; lanes16-31 K=k+2,k+3
//    B-frag (mirrored):            lanes0-15 hold rows k,k+1; hi lanes rows k+2,k+3
//    C (16x16 f32 layout): VGPR v -> M=v (lanes0-15) / M=v+8 (lanes16-31), N=lane%16
// ---------------------------------------------------------------
__global__ __launch_bounds__(256)
void gat_gemm_wmma(const float* __restrict__ x, const float* __restrict__ W,
                   float* __restrict__ h) {
    __shared__ float As[16 * IN_F];           // 8 KB A tile
    const int tid = threadIdx.x;
    const int m0  = blockIdx.x * 16;

    // cooperative A-tile load: 2048 floats / 256 threads = 2 x float4 each
    {
        const int base = tid * 8;
        const int row  = base >> 7;           // /128
        const int col  = base & 127;
        const float4* g =
            reinterpret_cast<const float4*>(x + (size_t)(m0 + row) * IN_F + col);
        float4 p0 = g[0];
        float4 p1 = g[1];
        *reinterpret_cast<float4*>(&As[row * IN_F + col])     = p0;
        *reinterpret_cast<float4*>(&As[row * IN_F + col + 4]) = p1;
    }
    __syncthreads();

    const int lane = tid & 31;
    const int head = tid >> 5;                 // 0..7
    const int l16  = lane & 15;
    const bool hi  = lane >= 16;
    const float* Wh = W + (size_t)head * IN_F * HEAD_DIM;  // W[head][k][d]

    v8f c = {};
#pragma unroll
    for (int k = 0; k < IN_F; k += 4) {
        const int kk = k + (hi ? 2 : 0);
        v2f a, b;
        a.x = As[l16 * IN_F + kk];
        a.y = As[l16 * IN_F + kk + 1];
        b.x = Wh[kk * HEAD_DIM + l16];
        b.y = Wh[(kk + 1) * HEAD_DIM + l16];
        c = __builtin_amdgcn_wmma_f32_16x16x4_f32(
                /*neg_a=*/false, a, /*neg_b=*/false, b,
                /*c_mod=*/(short)0, c, /*reuse_a=*/false, /*reuse_b=*/false);
    }

    const int rbase = hi ? 8 : 0;
#pragma unroll
    for (int v = 0; v < 8; ++v)
        h[(size_t)(m0 + rbase + v) * OUT_F + head * HEAD_DIM + l16] = c[v];
}

// ---------------------------------------------------------------
// 2) per-(node,head) attention logits: s = <h[n,h,:], a_{src/tgt}[h,:]>
// ---------------------------------------------------------------
__global__ void gat_node_scores(const float* __restrict__ h, const float* __restrict__ a,
                                float* __restrict__ s_src, float* __restrict__ s_tgt) {
    const int i = blockIdx.x * blockDim.x + threadIdx.x;
    if (i >= N_NODES * HEADS) return;
    const int hd = i & (HEADS - 1);
    const int n  = i >> 3;
    const float* hv = h + (size_t)n * OUT_F + hd * HEAD_DIM;
    const float* as = a + hd * 2 * HEAD_DIM;        // a[h][0:16]
    const float* at = as + HEAD_DIM;                // a[h][16:32]
    float ss = 0.0f, st = 0.0f;
#pragma unroll
    for (int d = 0; d < HEAD_DIM; ++d) {
        const float v = hv[d];
        ss += v * as[d];
        st += v * at[d];
    }
    s_src[i] = ss;
    s_tgt[i] = st;
}

// ---------------------------------------------------------------
// 3) per-head global max of leaky scores (LDS pre-reduce + global atomicMax)
// ---------------------------------------------------------------
__global__ __launch_bounds__(256)
void gat_edge_max(const int* __restrict__ src, const int* __restrict__ tgt,
                  const float* __restrict__ s_src, const float* __restrict__ s_tgt,
                  unsigned* __restrict__ gmax) {
    __shared__ unsigned lmax[HEADS];
    const int tid = threadIdx.x;
    if (tid < HEADS) lmax[tid] = 0u;
    __syncthreads();

    const int e = blockIdx.x * blockDim.x + tid;
    if (e < N_EDGES) {
        const int s = src[e], t = tgt[e];
#pragma unroll
        for (int hd = 0; hd < HEADS; ++hd) {
            const float sc = leaky(s_src[s * HEADS + hd] + s_tgt[t * HEADS + hd]);
            atomicMax(&lmax[hd], enc_f32(sc));
        }
    }
    __syncthreads();
    if (tid < HEADS) atomicMax(&gmax[tid], lmax[tid]);
}

// ---------------------------------------------------------------
// 4) softmax denominator: denom[t,h] += exp(clip(score - max_h))
// ---------------------------------------------------------------
__global__ __launch_bounds__(256)
void gat_edge_denom(const int* __restrict__ src, const int* __restrict__ tgt,
                    const float* __restrict__ s_src, const float* __restrict__ s_tgt,
                    const unsigned* __restrict__ gmax, float* __restrict__ denom) {
    const int e = blockIdx.x * blockDim.x + threadIdx.x;
    if (e >= N_EDGES) return;
    const int s = src[e], t = tgt[e];
#pragma unroll
    for (int hd = 0; hd < HEADS; ++hd) {
        const float mx = dec_f32(gmax[hd]);
        const float sc = leaky(s_src[s * HEADS + hd] + s_tgt[t * HEADS + hd]);
        const float ev = __expf(fminf(fmaxf(sc - mx, -20.0f), 20.0f));
        atomicAdd(&denom[t * HEADS + hd], ev);
    }
}

// ---------------------------------------------------------------
// 5) aggregation: out[t, c] += h[s, c] * attn(e, head(c))
//    256 thr = 2 edges x 128 channels; contiguous 512B gather + atomic scatter
// ---------------------------------------------------------------
__global__ __launch_bounds__(256)
void gat_edge_agg(const int* __restrict__ src, const int* __restrict__ tgt,
                  const float* __restrict__ s_src, const float* __restrict__ s_tgt,
                  const unsigned* __restrict__ gmax, const float* __restrict__ denom,
                  const float* __restrict__ h, float* __restrict__ out) {
    const long long gid = (long long)blockIdx.x * blockDim.x + threadIdx.x;
    const int e  = (int)(gid >> 7);       // /128
    const int c  = (int)(gid & 127);
    const int hd = c >> 4;
    if (e >= N_EDGES) return;
    const int s = src[e], t = tgt[e];
    const float mx = dec_f32(gmax[hd]);
    const float sc = leaky(s_src[s * HEADS + hd] + s_tgt[t * HEADS + hd]);
    const float ev = __expf(fminf(fmaxf(sc - mx, -20.0f), 20.0f));
    const float attn = ev / (denom[t * HEADS + hd] + EPS_F);
    atomicAdd(&out[(size_t)t * OUT_F + c], h[(size_t)s * OUT_F + c] * attn);
}

// ---------------------------------------------------------------
extern "C" void kernel_launch(void* const* d_in, const int* in_sizes, int n_in,
                              void* d_out, int out_size, void* d_ws, size_t ws_size,
                              hipStream_t stream) {
    (void)in_sizes; (void)n_in; (void)out_size; (void)ws_size;

    const float* x    = (const float*)d_in[0];
    const int*   ei   = (const int*)  d_in[1];
    const float* W    = (const float*)d_in[2];
    const float* a    = (const float*)d_in[3];
    const float* bias = (const float*)d_in[4];
    float* out = (float*)d_out;

    const int* src = ei;                // edge_index[0]
    const int* tgt = ei + N_EDGES;      // edge_index[1]

    // workspace layout (floats): h | s_src | s_tgt | denom | gmax  (~30.4 MB)
    float*    ws    = (float*)d_ws;
    float*    hbuf  = ws;                                   // 6,400,000
    float*    ssrc  = hbuf + (size_t)N_NODES * OUT_F;       //   400,000
    float*    stgt  = ssrc + (size_t)N_NODES * HEADS;       //   400,000
    float*    denom = stgt + (size_t)N_NODES * HEADS;       //   400,000
    unsigned* gmax  = (unsigned*)(denom + (size_t)N_NODES * HEADS);  // 8

    const int outN = N_NODES * OUT_F;
    gat_init<<<(outN + 255) / 256, 256, 0, stream>>>(out, bias, denom, gmax);

    gat_gemm_wmma<<<N_NODES / 16, 256, 0, stream>>>(x, W, hbuf);

    gat_node_scores<<<(N_NODES * HEADS + 255) / 256, 256, 0, stream>>>(hbuf, a, ssrc, stgt);

    gat_edge_max<<<(N_EDGES + 255) / 256, 256, 0, stream>>>(src, tgt, ssrc, stgt, gmax);

    gat_edge_denom<<<(N_EDGES + 255) / 256, 256, 0, stream>>>(src, tgt, ssrc, stgt, gmax, denom);

    const long long aggThreads = (long long)N_EDGES * OUT_F;   // 102.4M
    gat_edge_agg<<<(unsigned)((aggThreads + 255) / 256), 256, 0, stream>>>(
        src, tgt, ssrc, stgt, gmax, denom, hbuf, out);
}